// VP_lattice_22033182228961
// MI455X (gfx1250) — compile-verified
//
#include <hip/hip_runtime.h>
#include <math.h>

#define NUM_STEPS 1000
#define TBL (NUM_STEPS + 1)   /* 1001 */
#define EPS_F 1e-8f

#if defined(__gfx1250__) && __has_builtin(__builtin_amdgcn_global_load_async_to_lds_b32)
#define USE_ASYNC_LDS 1
#endif

__global__ __launch_bounds__(256) void vp_lattice_kernel(
    const float* __restrict__ lt,
    const float* __restrict__ l0,
    const float* __restrict__ latt,
    const float* __restrict__ alpha_bars,
    const float* __restrict__ betas,
    const float* __restrict__ sigmas,
    const float* __restrict__ zn,
    const int* __restrict__ tt,
    float* __restrict__ out,
    int B)
{
    __shared__ float s_beta[TBL];
    __shared__ float s_ab[TBL];
    __shared__ float s_sig[TBL];

    // ---- stage schedule tables into LDS (ASYNCcnt path on gfx1250) ----
    for (int idx = threadIdx.x; idx < TBL; idx += blockDim.x) {
#if USE_ASYNC_LDS
        __builtin_amdgcn_global_load_async_to_lds_b32(
            (int*)(betas + idx),      (int*)(s_beta + idx), 0, 0);
        __builtin_amdgcn_global_load_async_to_lds_b32(
            (int*)(alpha_bars + idx), (int*)(s_ab + idx),   0, 0);
        __builtin_amdgcn_global_load_async_to_lds_b32(
            (int*)(sigmas + idx),     (int*)(s_sig + idx),  0, 0);
#else
        s_beta[idx] = betas[idx];
        s_ab[idx]   = alpha_bars[idx];
        s_sig[idx]  = sigmas[idx];
#endif
    }
#if USE_ASYNC_LDS
#if __has_builtin(__builtin_amdgcn_s_wait_asynccnt)
    __builtin_amdgcn_s_wait_asynccnt(0);
#else
    asm volatile("s_wait_asynccnt 0" ::: "memory");
#endif
#endif
    __syncthreads();

    const float alpha_floor = 1.0f - s_beta[NUM_STEPS - 1];  // betas[-2] == betas[999]

    const int stride = (int)(blockDim.x * gridDim.x);
    for (int i = (int)(blockIdx.x * blockDim.x + threadIdx.x); i < B; i += stride) {
        // ------------- load 3x3 lattice, form M = A^T A in f64 -------------
        const float* Ap = latt + (size_t)i * 9;
        double d0 = Ap[0], d1 = Ap[1], d2 = Ap[2];
        double d3 = Ap[3], d4 = Ap[4], d5 = Ap[5];
        double d6 = Ap[6], d7 = Ap[7], d8 = Ap[8];

        double m00 = d0*d0 + d3*d3 + d6*d6;
        double m01 = d0*d1 + d3*d4 + d6*d7;
        double m02 = d0*d2 + d3*d5 + d6*d8;
        double m11 = d1*d1 + d4*d4 + d7*d7;
        double m12 = d1*d2 + d4*d5 + d7*d8;
        double m22 = d2*d2 + d5*d5 + d8*d8;

        // ------------- eigenvalues of symmetric M (trig closed form) -------------
        double q  = (m00 + m11 + m22) * (1.0 / 3.0);
        double p1 = m01*m01 + m02*m02 + m12*m12;
        double b00 = m00 - q, b11 = m11 - q, b22 = m22 - q;
        double p2 = b00*b00 + b11*b11 + b22*b22 + 2.0 * p1;
        double p  = sqrt(fmax(p2, 1e-300) * (1.0 / 6.0));
        double ip = 1.0 / p;
        double c00 = b00*ip, c11 = b11*ip, c22 = b22*ip;
        double c01 = m01*ip, c02 = m02*ip, c12 = m12*ip;
        double detB = c00*(c11*c22 - c12*c12)
                    - c01*(c01*c22 - c12*c02)
                    + c02*(c01*c12 - c11*c02);
        double r = 0.5 * detB;
        r = fmin(1.0, fmax(-1.0, r));
        double phi  = acos(r) * (1.0 / 3.0);
        double twop = 2.0 * p;
        double lam0 = q + twop * cos(phi);
        double lam2 = q + twop * cos(phi + 2.0943951023931953);  // + 2*pi/3
        double lam1 = 3.0 * q - lam0 - lam2;

        // singular values of A = sqrt(eigenvalues of M)
        double s0 = sqrt(fmax(lam0, 0.0));
        double s1 = sqrt(fmax(lam1, 0.0));
        double s2 = sqrt(fmax(lam2, 0.0));

        // ------------- H = sqrt(M) via invariants (no eigenvectors) -------------
        double IH   = s0 + s1 + s2;
        double IIH  = s0*s1 + s0*s2 + s1*s2;
        double IIIH = s0*s1*s2;
        double den  = fmax(IH*IIH - IIIH, 1e-300);   // = (s0+s1)(s0+s2)(s1+s2)
        double idn  = 1.0 / den;
        double cI   = IH * IIIH;
        double cM   = IH * IH - IIH;

        double M200 = m00*m00 + m01*m01 + m02*m02;
        double M201 = m00*m01 + m01*m11 + m02*m12;
        double M202 = m00*m02 + m01*m12 + m02*m22;
        double M211 = m01*m01 + m11*m11 + m12*m12;
        double M212 = m01*m02 + m11*m12 + m12*m22;
        double M222 = m02*m02 + m12*m12 + m22*m22;

        float v[6];
        v[0] = (float)((cI + cM*m00 - M200) * idn);  // L[0,0]
        v[1] = (float)((     cM*m01 - M201) * idn);  // L[0,1]
        v[2] = (float)((     cM*m02 - M202) * idn);  // L[0,2]
        v[3] = (float)((cI + cM*m11 - M211) * idn);  // L[1,1]
        v[4] = (float)((     cM*m12 - M212) * idn);  // L[1,2]
        v[5] = (float)((cI + cM*m22 - M222) * idn);  // L[2,2]

        // ------------- diffusion reverse step (f32, tables from LDS) -------------
        int t = tt[i];
        float beta_t    = s_beta[t];
        float alpha     = fmaxf(1.0f - beta_t, alpha_floor);
        float alpha_bar = s_ab[t];
        float sigma     = s_sig[t];
        float coef  = 1.0f / sqrtf(alpha + EPS_F);
        float scale = (1.0f - alpha) / sqrtf(1.0f - alpha_bar + EPS_F);
        bool  use_z = (t > 1);

        const float* ltp = lt + (size_t)i * 6;
        const float* l0p = l0 + (size_t)i * 6;
        const float* znp = zn + (size_t)i * 6;
        float*       op  = out + (size_t)i * 6;

#pragma unroll
        for (int k = 0; k < 6; ++k) {
            float l  = ltp[k];
            float pn = l - 0.5f * (v[k] + l0p[k]);
            float z  = use_z ? znp[k] : 0.0f;
            op[k] = coef * (l - scale * pn) + sigma * z;
        }
    }
}

extern "C" void kernel_launch(void* const* d_in, const int* in_sizes, int n_in,
                              void* d_out, int out_size, void* d_ws, size_t ws_size,
                              hipStream_t stream) {
    (void)n_in; (void)d_ws; (void)ws_size;
    const float* lt         = (const float*)d_in[0];
    const float* l0         = (const float*)d_in[1];
    const float* latt       = (const float*)d_in[2];
    const float* alpha_bars = (const float*)d_in[3];
    const float* betas      = (const float*)d_in[4];
    const float* sigmas     = (const float*)d_in[5];
    const float* zn         = (const float*)d_in[6];
    const int*   tt         = (const int*)d_in[7];
    float* out = (float*)d_out;

    int B = in_sizes[0] / 6;
    const int block = 256;
    const int rows_per_thread = 4;        // amortize the 12 KB LDS staging
    int grid = (B + block * rows_per_thread - 1) / (block * rows_per_thread);
    if (grid < 1) grid = 1;

    vp_lattice_kernel<<<grid, block, 0, stream>>>(lt, l0, latt, alpha_bars, betas,
                                                  sigmas, zn, tt, out, B);
}